// GraphSAGE_19636590477698
// MI455X (gfx1250) — compile-verified
//
#include <hip/hip_runtime.h>

// GraphSAGE (2-layer, mean aggregation) for MI455X / gfx1250.
// Exploits linearity: lin_l(mean_j x_j) == mean_j(lin_l x_j), so we project to
// 64-dim with WMMA GEMMs *before* the edge scatter, halving layer-1 edge traffic.
// GEMMs use native fp32 WMMA (v_wmma_f32_16x16x4_f32) -> f32-faithful math.
// Scatter uses hardware global_atomic_add_f32 (forced via inline asm).

typedef __attribute__((ext_vector_type(2))) float v2f;
typedef __attribute__((ext_vector_type(8))) float v8f;

#define HIDDEN 64

// Non-returning hardware fp32 atomic add (ISA §15.18.3 op 86). STOREcnt-tracked;
// implicitly drained at s_endpgm, visible to subsequent kernels via L2.
__device__ __forceinline__ void atomic_add_f32(float* p, float v) {
  asm volatile("global_atomic_add_f32 %0, %1, off" : : "v"(p), "v"(v) : "memory");
}

__global__ void zero_kernel(float* __restrict__ p, int n) {
  int t = blockIdx.x * blockDim.x + threadIdx.x;
  if (t < n) p[t] = 0.0f;
}

// cnt[dst] += 1 for every edge (fp32 degrees; reused by both layers)
__global__ void degree_kernel(const int* __restrict__ ei, float* __restrict__ cnt, int E) {
  int e = blockIdx.x * blockDim.x + threadIdx.x;
  if (e < E) {
    int d = ei[E + e];  // edge_index[1][e] = dst
    atomic_add_f32(&cnt[d], 1.0f);
  }
}

// Dual GEMM: OutL = X[nrows,K] * Wl[64,K]^T ; OutR = X * Wr^T.
// One wave -> one 16-row tile x full 64 cols, for BOTH weights (A-fragment reuse).
// 4 waves / 128-thread block -> 64 rows per block.
// Per k-step: gather all 9 fragments first so the backend can clause the loads
// and stagger s_wait_loadcnt instead of a load->wait0->wmma lockstep.
template <int K>
__global__ void sage_dual_gemm(const float* __restrict__ X,
                               const float* __restrict__ Wl,
                               const float* __restrict__ Wr,
                               float* __restrict__ OutL,
                               float* __restrict__ OutR,
                               int nrows) {
  const int lane  = threadIdx.x & 31;
  const int wave  = threadIdx.x >> 5;
  const int row0  = blockIdx.x * 64 + wave * 16;
  if (row0 >= nrows) return;

  const int m     = lane & 15;   // matrix row (A) / col (B,C,D) owned by this lane
  const int khalf = lane >> 4;   // which K-pair this lane holds (ISA 16x4 f32 layout)

  v8f accL[4] = {};
  v8f accR[4] = {};

  const float* xrow = X + (size_t)(row0 + m) * K + khalf * 2;

  for (int k0 = 0; k0 < K; k0 += 4) {
    // ---- load phase: A fragment + 8 B fragments (clause-friendly) ----
    const v2f a = *reinterpret_cast<const v2f*>(xrow + k0);
    v2f bl[4], br[4];
#pragma unroll
    for (int t = 0; t < 4; ++t) {
      const int n = t * 16 + m;  // output column; B = W^T so this reads W[n][k..k+1]
      bl[t] = *reinterpret_cast<const v2f*>(Wl + (size_t)n * K + k0 + khalf * 2);
      br[t] = *reinterpret_cast<const v2f*>(Wr + (size_t)n * K + k0 + khalf * 2);
    }
    // ---- math phase: 8 WMMAs consuming the staged fragments ----
#pragma unroll
    for (int t = 0; t < 4; ++t) {
      accL[t] = __builtin_amdgcn_wmma_f32_16x16x4_f32(
          false, a, false, bl[t], (short)0, accL[t], false, false);
      accR[t] = __builtin_amdgcn_wmma_f32_16x16x4_f32(
          false, a, false, br[t], (short)0, accR[t], false, false);
    }
  }

  // C/D layout: VGPR r <-> row (r + 8*khalf), col = t*16 + m
#pragma unroll
  for (int t = 0; t < 4; ++t) {
#pragma unroll
    for (int r = 0; r < 8; ++r) {
      const size_t row = (size_t)(row0 + khalf * 8 + r);
      const int    col = t * 16 + m;
      OutL[row * HIDDEN + col] = accL[t][r];
      OutR[row * HIDDEN + col] = accR[t][r];
    }
  }
}

// sums[dst][d] += feat[src][d] over all edges. 32 lanes per edge: each lane
// gathers a float2 (b64, coalesced 256B/edge) and issues two hardware atomics.
__global__ void scatter_accum(const int* __restrict__ ei, const float* __restrict__ feat,
                              float* __restrict__ sums, int E) {
  long long t = (long long)blockIdx.x * blockDim.x + threadIdx.x;
  int e = (int)(t >> 5);
  int d = (int)(t & 31) * 2;
  if (e < E) {
    int s  = ei[e];       // src
    int dd = ei[E + e];   // dst
    const v2f f = *reinterpret_cast<const v2f*>(feat + (size_t)s * HIDDEN + d);
    float* p = sums + (size_t)dd * HIDDEN + d;
    atomic_add_f32(p,     f.x);
    atomic_add_f32(p + 1, f.y);
  }
}

// h = relu(S/max(cnt,1) + b + R)
__global__ void epilogue_relu(const float* __restrict__ S, const float* __restrict__ cnt,
                              const float* __restrict__ b, const float* __restrict__ R,
                              float* __restrict__ out, int n64) {
  int t = blockIdx.x * blockDim.x + threadIdx.x;
  if (t < n64) {
    int i = t >> 6, d = t & 63;
    float v = S[t] / fmaxf(cnt[i], 1.0f) + b[d] + R[t];
    out[t] = fmaxf(v, 0.0f);
  }
}

// out += S/max(cnt,1) + b   (out already holds h @ W2_r^T)
__global__ void epilogue_add(const float* __restrict__ S, const float* __restrict__ cnt,
                             const float* __restrict__ b, float* __restrict__ out, int n64) {
  int t = blockIdx.x * blockDim.x + threadIdx.x;
  if (t < n64) {
    int i = t >> 6, d = t & 63;
    out[t] = S[t] / fmaxf(cnt[i], 1.0f) + b[d] + out[t];
  }
}

extern "C" void kernel_launch(void* const* d_in, const int* in_sizes, int n_in,
                              void* d_out, int out_size, void* d_ws, size_t ws_size,
                              hipStream_t stream) {
  const float* x   = (const float*)d_in[0];
  const int*   ei  = (const int*)d_in[1];
  const float* W1l = (const float*)d_in[2];
  const float* b1l = (const float*)d_in[3];
  const float* W1r = (const float*)d_in[4];
  const float* W2l = (const float*)d_in[5];
  const float* b2l = (const float*)d_in[6];
  const float* W2r = (const float*)d_in[7];
  float* out = (float*)d_out;

  const int IN  = 128;
  const int E   = in_sizes[1] / 2;
  const int N   = in_sizes[0] / IN;
  const int n64 = N * HIDDEN;

  // workspace layout (floats): cnt[N] | S[N*64] | A[N*64] | B[N*64]  (~31 MB)
  float* cnt = (float*)d_ws;
  float* S   = cnt + N;
  float* A   = S + n64;   // xl -> h
  float* B   = A + n64;   // xr -> hl

  const int TB = 256;
  const int gemmBlocks = (N + 63) / 64;
  const long long sth  = (long long)E * 32;   // 32 lanes per edge
  const int scatBlocks = (int)((sth + TB - 1) / TB);

  // 1) zero cnt + S (contiguous)
  zero_kernel<<<(N + n64 + TB - 1) / TB, TB, 0, stream>>>(cnt, N + n64);
  // 2) degrees
  degree_kernel<<<(E + TB - 1) / TB, TB, 0, stream>>>(ei, cnt, E);
  // 3) layer 1 projections: A = x@W1_l^T, B = x@W1_r^T   (WMMA, K=128)
  sage_dual_gemm<128><<<gemmBlocks, 128, 0, stream>>>(x, W1l, W1r, A, B, N);
  // 4) S[dst] += A[src] over edges
  scatter_accum<<<scatBlocks, TB, 0, stream>>>(ei, A, S, E);
  // 5) A = h = relu(S/cnt + b1 + B)
  epilogue_relu<<<(n64 + TB - 1) / TB, TB, 0, stream>>>(S, cnt, b1l, B, A, n64);
  // 6) re-zero S
  zero_kernel<<<(n64 + TB - 1) / TB, TB, 0, stream>>>(S, n64);
  // 7) layer 2 projections: B = h@W2_l^T, out = h@W2_r^T   (WMMA, K=64)
  sage_dual_gemm<64><<<gemmBlocks, 128, 0, stream>>>(A, W2l, W2r, B, out, N);
  // 8) S[dst] += B[src] over edges
  scatter_accum<<<scatBlocks, TB, 0, stream>>>(ei, B, S, E);
  // 9) out += S/cnt + b2
  epilogue_add<<<(n64 + TB - 1) / TB, TB, 0, stream>>>(S, cnt, b2l, out, n64);
}